// JointDGMRF_53893249630423
// MI455X (gfx1250) — compile-verified
//
#include <hip/hip_runtime.h>

// DGMRF forward, MI455X (gfx1250, wave32).
// Memory/atomic-bound scatter-gather. Key choices for MI455X:
//  - state kept node-major [N,4]: 1x global_load_b128 gather per edge and
//    4 same-cacheline global_atomic_add_f32 per edge (L2-resident aggr),
//  - edge indices (64MB/layer) double-buffered global->LDS with CDNA5 async
//    copies (ASYNCcnt + s_wait_asynccnt); each lane reads back only its own
//    16B slot -> no barriers,
//  - edge weight exp((p-1)*log_deg[dst]) factored out of the edge loop and
//    applied once per node in the fused update (which also re-zeroes aggr).

#define TILE_EDGES 1024          // 256 threads * 4 edges
#define SCATTER_THREADS 256
#define T_CH 4

// ---- CDNA5 async global->LDS copy, 16B per lane, tracked by ASYNCcnt ----
__device__ __forceinline__ void async_copy_b128(unsigned lds_off, const void* gptr) {
    unsigned long long ga = (unsigned long long)(uintptr_t)gptr;
    asm volatile("global_load_async_to_lds_b128 %0, %1, off"
                 :: "v"(lds_off), "v"(ga) : "memory");
}
__device__ __forceinline__ void wait_async_le2() {
    asm volatile("s_wait_asynccnt 2" ::: "memory");
}
__device__ __forceinline__ void wait_async_0() {
    asm volatile("s_wait_asynccnt 0" ::: "memory");
}

// ---- degree count: int atomics into cnt[] ----
__global__ void dgmrf_count(const int* __restrict__ src, int* __restrict__ cnt,
                            long long E) {
    long long i = ((long long)blockIdx.x * blockDim.x + threadIdx.x) * 4;
    if (i + 3 < E) {
        int4 s = *(const int4*)(src + i);
        atomicAdd(&cnt[s.x], 1);
        atomicAdd(&cnt[s.y], 1);
        atomicAdd(&cnt[s.z], 1);
        atomicAdd(&cnt[s.w], 1);
    } else {
        for (; i < E; ++i) atomicAdd(&cnt[src[i]], 1);
    }
}

// ---- cnt (int) -> log_deg (float), in place ----
__global__ void dgmrf_logdeg(float* __restrict__ buf, int N) {
    int n = blockIdx.x * blockDim.x + threadIdx.x;
    if (n < N) {
        int c = ((const int*)buf)[n];
        buf[n] = logf((float)c);
    }
}

// ---- [T,N] -> [N,4] transpose (coalesced reads per plane, b128 stores) ----
__global__ void dgmrf_transpose_in(const float* __restrict__ x,
                                   float* __restrict__ xT, int N) {
    int n = blockIdx.x * blockDim.x + threadIdx.x;
    if (n < N) {
        float4 v;
        v.x = x[(size_t)0 * N + n];
        v.y = x[(size_t)1 * N + n];
        v.z = x[(size_t)2 * N + n];
        v.w = x[(size_t)3 * N + n];
        *(float4*)(xT + (size_t)n * 4) = v;
    }
}

// ---- scatter: aggrT[dst[e]*4+t] += xT[src[e]*4+t] (weight factored out) ----
__global__ void __launch_bounds__(SCATTER_THREADS)
dgmrf_scatter(const float* __restrict__ xT, const int* __restrict__ src,
              const int* __restrict__ dst, float* __restrict__ aggrT,
              long long E, int numTiles) {
    __shared__ int s_src[2][TILE_EDGES];
    __shared__ int s_dst[2][TILE_EDGES];
    const int tid  = threadIdx.x;
    const int slot = tid * 4;

    unsigned lsrc[2], ldst[2];
    lsrc[0] = (unsigned)(uintptr_t)&s_src[0][slot];
    lsrc[1] = (unsigned)(uintptr_t)&s_src[1][slot];
    ldst[0] = (unsigned)(uintptr_t)&s_dst[0][slot];
    ldst[1] = (unsigned)(uintptr_t)&s_dst[1][slot];

    int tile = blockIdx.x;
    if (tile >= numTiles) return;

    {   // prime the pipeline
        long long b  = (long long)tile * TILE_EDGES + slot;
        long long cb = (b < E) ? b : 0;   // clamp: OOB lanes load junk, never used
        async_copy_b128(lsrc[0], src + cb);
        async_copy_b128(ldst[0], dst + cb);
    }

    int buf = 0;
    for (; tile < numTiles; tile += gridDim.x) {
        int next = tile + gridDim.x;
        if (next < numTiles) {
            long long nb = (long long)next * TILE_EDGES + slot;
            long long cb = (nb < E) ? nb : 0;
            async_copy_b128(lsrc[buf ^ 1], src + cb);
            async_copy_b128(ldst[buf ^ 1], dst + cb);
            wait_async_le2();             // current tile's 2 copies are done
        } else {
            wait_async_0();
        }

        long long base = (long long)tile * TILE_EDGES + slot;
        if (base < E) {                   // E % 4 == 0 -> full quad or nothing
            int4 sv = *(const int4*)&s_src[buf][slot];   // ds_load_b128
            int4 dv = *(const int4*)&s_dst[buf][slot];
            int ss[4] = {sv.x, sv.y, sv.z, sv.w};
            int dd[4] = {dv.x, dv.y, dv.z, dv.w};
#pragma unroll
            for (int i = 0; i < 4; ++i) {
                float4 xv = *(const float4*)(xT + (size_t)ss[i] * 4); // 1 line
                float* ad = aggrT + (size_t)dd[i] * 4;                // 1 line
                atomicAdd(ad + 0, xv.x);
                atomicAdd(ad + 1, xv.y);
                atomicAdd(ad + 2, xv.z);
                atomicAdd(ad + 3, xv.w);
            }
        }
        buf ^= 1;
    }
}

// ---- fused node update in [N,4]; re-zeroes aggr; last layer writes [T,N] ----
__global__ void dgmrf_update(const float* __restrict__ xinT,
                             float* __restrict__ aggrT,
                             const float* __restrict__ log_deg,
                             const float* __restrict__ alpha1,
                             const float* __restrict__ gammap,
                             const float* __restrict__ biasp,
                             float* __restrict__ xout,
                             int N, int layer, int toChannelMajor) {
    int n = blockIdx.x * blockDim.x + threadIdx.x;
    if (n >= N) return;

    const float a  = alpha1[layer];
    const float g  = gammap[layer];
    const float b  = biasp[layer];
    const float p  = 1.0f / (1.0f + expf(-g));   // sigmoid(gamma)
    const float sw = expf(a);                    // self weight
    const float nw = sw * tanhf(a);              // neighbor weight

    const float ld  = log_deg[n];
    const float wsc = sw * expf(p * ld);
    const float wnc = nw * expf((p - 1.0f) * ld);

    float4  xv = *(const float4*)(xinT + (size_t)n * 4);
    float4* ap = (float4*)(aggrT + (size_t)n * 4);
    float4  av = *ap;
    *ap = make_float4(0.f, 0.f, 0.f, 0.f);       // ready for next layer

    float4 o;
    o.x = wsc * xv.x + wnc * av.x + b;
    o.y = wsc * xv.y + wnc * av.y + b;
    o.z = wsc * xv.z + wnc * av.z + b;
    o.w = wsc * xv.w + wnc * av.w + b;

    if (toChannelMajor) {                        // final layer -> d_out [T,N]
        xout[(size_t)0 * N + n] = o.x;           // 4 coalesced store streams
        xout[(size_t)1 * N + n] = o.y;
        xout[(size_t)2 * N + n] = o.z;
        xout[(size_t)3 * N + n] = o.w;
    } else {
        *(float4*)(xout + (size_t)n * 4) = o;
    }
}

extern "C" void kernel_launch(void* const* d_in, const int* in_sizes, int n_in,
                              void* d_out, int out_size, void* d_ws, size_t ws_size,
                              hipStream_t stream) {
    const float* x      = (const float*)d_in[0];   // [T, N]
    const float* alpha1 = (const float*)d_in[1];   // [L]
    const float* gam    = (const float*)d_in[2];   // [L]
    const float* bias   = (const float*)d_in[3];   // [L]
    const int*   edge   = (const int*)d_in[4];     // [2, E]

    const int       T = T_CH;
    const int       L = in_sizes[1];
    const long long E = (long long)in_sizes[4] / 2;
    const int       N = in_sizes[0] / T;

    const int* srcp = edge;
    const int* dstp = edge + E;

    // workspace: [log_deg N] [aggrT N*4] [xT0 N*4] [xT1 N*4]   (node-major)
    char*  wsb     = (char*)d_ws;
    float* log_deg = (float*)wsb;
    size_t off     = (((size_t)N * 4) + 255) & ~(size_t)255;
    float* aggrT   = (float*)(wsb + off);
    float* xT0     = aggrT + (size_t)T * N;
    float* xT1     = xT0   + (size_t)T * N;

    hipMemsetAsync(log_deg, 0, (size_t)N * 4, stream);
    hipMemsetAsync(aggrT,   0, (size_t)T * N * 4, stream);

    long long cthreads = (E + 3) / 4;
    dgmrf_count<<<(unsigned)((cthreads + 255) / 256), 256, 0, stream>>>(
        srcp, (int*)log_deg, E);
    dgmrf_logdeg<<<(N + 255) / 256, 256, 0, stream>>>(log_deg, N);

    const int nblocks = (N + 255) / 256;
    dgmrf_transpose_in<<<nblocks, 256, 0, stream>>>(x, xT0, N);

    const int numTiles = (int)((E + TILE_EDGES - 1) / TILE_EDGES);
    const int sblocks  = numTiles < 2048 ? numTiles : 2048;

    for (int l = 0; l < L; ++l) {
        const float* xinT = (l & 1) ? xT1 : xT0;
        const int    last = (l == L - 1);
        float*       xo   = last ? (float*)d_out : ((l & 1) ? xT0 : xT1);
        dgmrf_scatter<<<sblocks, SCATTER_THREADS, 0, stream>>>(
            xinT, srcp, dstp, aggrT, E, numTiles);
        dgmrf_update<<<nblocks, 256, 0, stream>>>(
            xinT, aggrT, log_deg, alpha1, gam, bias, xo, N, l, last);
    }
}